// GradientMask_48395691491657
// MI455X (gfx1250) — compile-verified
//
#include <hip/hip_runtime.h>
#include <math.h>

#define IMG_H 512
#define IMG_W 512
#define NIMG  128        // b*c*n = 8*1*16
#define STRIP 128        // rows per block
#define TILE  16         // cols per block / rows per wave-tile
#define GROWS 140        // STRIP + 6 + 6 halo
#define GCOLS 22         // TILE + 3 + 3 halo
#define GSTR  24         // padded LDS stride for gradient

typedef __attribute__((ext_vector_type(2))) float v2f;
typedef __attribute__((ext_vector_type(8))) float v8f;

// ky normalization: 1 / sum_j exp(-0.005*(j-6)^2), sigma=10, 13 taps
#define KY_NORM 0.08234397f

__device__ __forceinline__ float ldx(const float* __restrict__ X, int r, int c) {
  // Sobel stage uses zero padding (jax conv pad=(1,1) with 0)
  return ((unsigned)r < (unsigned)IMG_H && (unsigned)c < (unsigned)IMG_W)
             ? X[r * IMG_W + c]
             : 0.0f;
}

__device__ __forceinline__ int refl(int v) {
  // jnp.pad 'reflect' (no edge repeat): -1 -> 1, 512 -> 510
  v = (v < 0) ? -v : v;
  return (v > IMG_H - 1) ? (2 * (IMG_H - 1) - v) : v;
}

// Banded-matrix element: ky[j] if j in [0,12] else 0 (branchless, no LDS)
__device__ __forceinline__ float ky_band(int j) {
  float t = (float)(j - 6);
  float w = __expf(-0.005f * t * t) * KY_NORM;
  return ((unsigned)j <= 12u) ? w : 0.0f;
}

// Computes the fully blurred gradient for this wave's 16x16 output tile.
// Result layout = WMMA 16x16 f32 C/D striping:
//   d[v] -> row h0 + 16*wave + v + 8*(lane>=16), col w0 + (lane&15)
__device__ __forceinline__ v8f blur_tile(const float* __restrict__ X, int h0,
                                         int w0, float* __restrict__ sG,
                                         float* __restrict__ sHb) {
  const int tid = threadIdx.x;

  // Stage 0: warm WGP$/L2 for this block's halo region (global_prefetch_b8).
  // The full input (134 MB) fits in the 192 MB L2, so pass 2 re-reads at L2 BW.
  for (int t = tid; t < GROWS + 2; t += 256) {
    int r = h0 - 7 + t;
    if ((unsigned)r < (unsigned)IMG_H) {
      const float* p = X + (size_t)r * IMG_W + ((w0 >= 4) ? (w0 - 4) : 0);
      __builtin_prefetch(p, 0, 1);
      __builtin_prefetch(p + 20, 0, 1);  // cover tail of the 24-col halo row
    }
  }

  // Stage 1: gradient magnitude over the halo region (140 rows x 22 cols)
  for (int t = tid; t < GROWS * GCOLS; t += 256) {
    int lr = t / GCOLS, lc = t - lr * GCOLS;
    int gr = h0 - 6 + lr, gc = w0 - 3 + lc;
    float gv = ldx(X, gr + 1, gc) - ldx(X, gr - 1, gc);
    float gh = ldx(X, gr, gc + 1) - ldx(X, gr, gc - 1);
    sG[lr * GSTR + lc] = sqrtf(gv * gv + gh * gh + 1e-6f);
  }
  __syncthreads();

  // Stage 2: horizontal 7-tap blur, reflect padding (rows stored reflected).
  // Taps are compile-time literals after full unroll.
  const float kx[7] = {0.13930901f, 0.14283564f, 0.14499434f, 0.14572111f,
                       0.14499434f, 0.14283564f, 0.13930901f};
  for (int t = tid; t < GROWS * TILE; t += 256) {
    int lr = t >> 4, j = t & 15;
    int lrr = refl(h0 - 6 + lr) - (h0 - 6);  // reflected row, local G index
    float s = 0.0f;
#pragma unroll
    for (int i = 0; i < 7; ++i) {
      int gc = refl(w0 + j - 3 + i);         // reflect columns at image edge
      s += kx[i] * sG[lrr * GSTR + (gc - (w0 - 3))];
    }
    sHb[lr * TILE + j] = s;
  }
  __syncthreads();

  // Stage 3: vertical 13-tap blur as banded matmul on the matrix unit.
  // out[M] = sum_{r=M..M+12} ky[r-M] * Hb[rowbase + r];  K = 28, 7 chunks of 4.
  const int lane = tid & 31;
  const int wv = tid >> 5;     // wave -> 16-row sub-tile
  const int half = lane >> 4;  // hi lanes hold K+2 (A/B) and M+8 (D)
  const int l16 = lane & 15;

  v8f acc = {};
#pragma unroll
  for (int kb = 0; kb < 28; kb += 4) {
    int r0 = kb + 2 * half;    // this lane's K indices: r0, r0+1
    int j0 = r0 - l16;         // banded A: ky tap index for row M=l16
    v2f a, b;
    a.x = ky_band(j0);         // pure VALU/trans, co-executes with WMMA
    a.y = ky_band(j0 + 1);
    b.x = sHb[(16 * wv + r0) * TILE + l16];
    b.y = sHb[(16 * wv + r0 + 1) * TILE + l16];
    acc = __builtin_amdgcn_wmma_f32_16x16x4_f32(false, a, false, b, (short)0,
                                                acc, false, false);
  }
  return acc;
}

__global__ __launch_bounds__(256) void gm_init(unsigned* __restrict__ mm) {
  int i = threadIdx.x;
  if (i < NIMG) {
    mm[i] = 0x7F800000u;   // +inf  (blur > 0, so float bits are monotonic)
    mm[NIMG + i] = 0u;     // 0.0f
  }
}

__global__ __launch_bounds__(256) void gm_pass1(const float* __restrict__ x,
                                                unsigned* __restrict__ mm) {
  __shared__ float sG[GROWS * GSTR];
  __shared__ float sHb[GROWS * TILE];
  const int tid = threadIdx.x;

  const int img = blockIdx.z;
  const int h0 = blockIdx.y * STRIP;
  const int w0 = blockIdx.x * TILE;
  const float* X = x + (size_t)img * IMG_H * IMG_W;

  v8f d = blur_tile(X, h0, w0, sG, sHb);

  float mn = d[0], mx = d[0];
#pragma unroll
  for (int v = 1; v < 8; ++v) {
    mn = fminf(mn, d[v]);
    mx = fmaxf(mx, d[v]);
  }
#pragma unroll
  for (int off = 16; off > 0; off >>= 1) {
    mn = fminf(mn, __shfl_xor(mn, off, 32));
    mx = fmaxf(mx, __shfl_xor(mx, off, 32));
  }
  if ((tid & 31) == 0) {
    atomicMin(&mm[img], __float_as_uint(mn));
    atomicMax(&mm[NIMG + img], __float_as_uint(mx));
  }
}

__global__ __launch_bounds__(256) void gm_pass2(const float* __restrict__ x,
                                                const unsigned* __restrict__ mm,
                                                float* __restrict__ out) {
  __shared__ float sG[GROWS * GSTR];
  __shared__ float sHb[GROWS * TILE];
  const int tid = threadIdx.x;

  const int img = blockIdx.z;
  const int h0 = blockIdx.y * STRIP;
  const int w0 = blockIdx.x * TILE;
  const float* X = x + (size_t)img * IMG_H * IMG_W;

  v8f d = blur_tile(X, h0, w0, sG, sHb);

  const float mn = __uint_as_float(mm[img]);
  const float mx = __uint_as_float(mm[NIMG + img]);
  const float inv = 1.0f / (mx - mn);

  const int lane = tid & 31;
  const int wv = tid >> 5;
  const int half = lane >> 4;
  const int l16 = lane & 15;
  const size_t base = (size_t)img * IMG_H * IMG_W;
  const size_t halfOut = (size_t)NIMG * IMG_H * IMG_W;

#pragma unroll
  for (int v = 0; v < 8; ++v) {
    int row = h0 + 16 * wv + v + 8 * half;
    int col = w0 + l16;
    float norm = (d[v] - mn) * inv;
    float e = (norm < 0.1f) ? 1.0f : 0.0f;
    size_t idx = base + (size_t)row * IMG_W + col;
    out[idx] = e;                   // maskEdge
    out[halfOut + idx] = 1.0f - e;  // maskNonEdge
  }
}

extern "C" void kernel_launch(void* const* d_in, const int* in_sizes, int n_in,
                              void* d_out, int out_size, void* d_ws,
                              size_t ws_size, hipStream_t stream) {
  (void)in_sizes; (void)n_in; (void)out_size; (void)ws_size;
  const float* x = (const float*)d_in[0];
  float* out = (float*)d_out;
  unsigned* mm = (unsigned*)d_ws;  // [128] min bits, [128] max bits

  gm_init<<<dim3(1), dim3(256), 0, stream>>>(mm);
  dim3 grid(IMG_W / TILE, IMG_H / STRIP, NIMG);
  gm_pass1<<<grid, dim3(256), 0, stream>>>(x, mm);
  gm_pass2<<<grid, dim3(256), 0, stream>>>(x, mm, out);
}